// Decoder_64020782514981
// MI455X (gfx1250) — compile-verified
//
#include <hip/hip_runtime.h>
#include <hip/hip_bf16.h>

#define HIDF 256
#define OUTF 128
#define MT   5          // row tiles per wave; N = 20000 = 250 * (16*MT)

typedef float v2f __attribute__((ext_vector_type(2)));
typedef float v8f __attribute__((ext_vector_type(8)));

// ---------------- degree / norm ----------------

__global__ void k_set_one(float* __restrict__ deg, int n) {
    int i = blockIdx.x * blockDim.x + threadIdx.x;
    if (i < n) deg[i] = 1.0f;               // self-loop contributes 1
}

__global__ void k_degree(const long long* __restrict__ dst,
                         float* __restrict__ deg, int nE) {
    int e = blockIdx.x * blockDim.x + threadIdx.x;
    if (e < nE) atomicAdd(&deg[(int)dst[e]], 1.0f);
}

__global__ void k_rsqrt(float* __restrict__ d, int n) {
    int i = blockIdx.x * blockDim.x + threadIdx.x;
    if (i < n) d[i] = rsqrtf(d[i]);         // deg >= 1 always (self-loops)
}

// ---------------- fp32 WMMA GEMM: out[N x FOUT] = A[N x 256] * W[256 x FOUT] ----------------
// One wave per (16*MT) x 16 output tile. K stepped by 4 using V_WMMA_F32_16X16X4_F32.
// A-frag (16x4 f32, 2 VGPRs/lane): lane m=l&15, kh=2*(l>>4): {A[m][k+kh], A[m][k+kh+1]}
// B-frag (4x16 f32, 2 VGPRs/lane): lane n=l&15, kh=2*(l>>4): {W[k+kh][n], W[k+kh+1][n]}
//   -> B loaded once per k-step, reused by all MT row tiles (amortized vmem).
// C/D  (16x16 f32, 8 VGPRs/lane): VGPR r -> row r + 8*(l>>4), col l&15
template <int FOUT>
__global__ void k_gemm_wmma(const float* __restrict__ A,
                            const float* __restrict__ W,
                            float* __restrict__ out) {
    const int lane = threadIdx.x;             // 0..31, wave32
    const int m0   = blockIdx.x * (16 * MT);  // N % (16*MT) == 0
    const int n0   = blockIdx.y * 16;
    const int mA   = lane & 15;
    const int kh   = (lane >> 4) * 2;         // 0 or 2
    const int nB   = lane & 15;

    const float* Arow[MT];
    #pragma unroll
    for (int t = 0; t < MT; ++t)
        Arow[t] = A + (size_t)(m0 + 16 * t + mA) * HIDF;
    const float* Wcol = W + (size_t)n0 + nB;

    v8f acc[MT];
    #pragma unroll
    for (int t = 0; t < MT; ++t) acc[t] = (v8f){};

    #pragma unroll 4
    for (int k = 0; k < HIDF; k += 4) {
        v2f b;
        b.x = Wcol[(size_t)(k + kh) * FOUT];
        b.y = Wcol[(size_t)(k + kh + 1) * FOUT];
        #pragma unroll
        for (int t = 0; t < MT; ++t) {
            v2f a = *(const v2f*)(Arow[t] + k + kh);
            acc[t] = __builtin_amdgcn_wmma_f32_16x16x4_f32(false, a, false, b,
                                                           (short)0, acc[t],
                                                           false, false);
        }
    }

    const int rbase = (lane >> 4) * 8;
    const int c     = lane & 15;
    #pragma unroll
    for (int t = 0; t < MT; ++t) {
        float* orow = out + (size_t)(m0 + 16 * t + rbase) * FOUT + n0 + c;
        #pragma unroll
        for (int r = 0; r < 8; ++r)
            orow[(size_t)r * FOUT] = acc[t][r];
    }
}

// ---------------- aggregation init: agg[i][j] = b[j] + xw[i][j]*dinv[i]^2 ----------------
template <int F>
__global__ void k_init_agg(const float* __restrict__ xw,
                           const float* __restrict__ dinv,
                           const float* __restrict__ bias,
                           float* __restrict__ agg, int n) {
    int g = blockIdx.x * blockDim.x + threadIdx.x;   // over n*F
    if (g >= n * F) return;
    int row = g / F;
    int j   = g - row * F;
    float di = dinv[row];
    agg[g] = bias[j] + xw[g] * di * di;
}

// ---------------- edge scatter: agg[dst] += xw[src] * dinv[src]*dinv[dst] ----------------
// F/4 lanes per edge, float4 per lane, f32 atomics (L2-resident).
template <int F>
__global__ void k_scatter(const long long* __restrict__ src,
                          const long long* __restrict__ dst,
                          const float* __restrict__ dinv,
                          const float* __restrict__ xw,
                          float* __restrict__ agg, int nE) {
    const int TPE = F / 4;
    int g = blockIdx.x * blockDim.x + threadIdx.x;
    int e = g / TPE;
    if (e >= nE) return;
    int j = (g - e * TPE) * 4;
    int s = (int)src[e];
    int d = (int)dst[e];
    float nrm = dinv[s] * dinv[d];
    const float4 v = *(const float4*)(xw + (size_t)s * F + j);
    float* o = agg + (size_t)d * F + j;
    atomicAdd(o + 0, v.x * nrm);
    atomicAdd(o + 1, v.y * nrm);
    atomicAdd(o + 2, v.z * nrm);
    atomicAdd(o + 3, v.w * nrm);
}

// ---------------- in-place ReLU (float4) ----------------
__global__ void k_relu4(float* __restrict__ h, int n4) {
    int i = blockIdx.x * blockDim.x + threadIdx.x;
    if (i >= n4) return;
    float4 v = ((float4*)h)[i];
    v.x = fmaxf(v.x, 0.0f); v.y = fmaxf(v.y, 0.0f);
    v.z = fmaxf(v.z, 0.0f); v.w = fmaxf(v.w, 0.0f);
    ((float4*)h)[i] = v;
}

extern "C" void kernel_launch(void* const* d_in, const int* in_sizes, int n_in,
                              void* d_out, int out_size, void* d_ws, size_t ws_size,
                              hipStream_t stream) {
    const float*     x  = (const float*)d_in[0];
    const long long* ei = (const long long*)d_in[1];     // int64 [2, E]
    const float*     W1 = (const float*)d_in[2];
    const float*     b1 = (const float*)d_in[3];
    const float*     W2 = (const float*)d_in[4];
    const float*     b2 = (const float*)d_in[5];
    const float*     W3 = (const float*)d_in[6];
    const float*     b3 = (const float*)d_in[7];
    float*           out = (float*)d_out;

    const int N = in_sizes[0] / HIDF;      // 20000
    const int E = in_sizes[1] / 2;         // 320000
    const long long* esrc = ei;
    const long long* edst = ei + E;

    // workspace: dinv[N] (padded), bufA[N*256], bufB[N*256]
    float* dinv = (float*)d_ws;
    float* bufA = dinv + ((N + 63) & ~63);
    float* bufB = bufA + (size_t)N * HIDF;

    const int T = 256;

    // --- normalization ---
    k_set_one<<<(N + T - 1) / T, T, 0, stream>>>(dinv, N);
    k_degree<<<(E + T - 1) / T, T, 0, stream>>>(edst, dinv, E);
    k_rsqrt<<<(N + T - 1) / T, T, 0, stream>>>(dinv, N);

    const dim3 gemmBlk(32, 1, 1);
    const dim3 gemmGrd1(N / (16 * MT), HIDF / 16, 1);   // 250 x 16
    const dim3 gemmGrd3(N / (16 * MT), OUTF / 16, 1);   // 250 x 8

    const int nElemH = N * HIDF;
    const int nElemO = N * OUTF;
    const int scatBlocksH = (E * (HIDF / 4) + T - 1) / T;
    const int scatBlocksO = (E * (OUTF / 4) + T - 1) / T;

    // --- layer 1: h1 = relu(agg(x @ W1) + b1) ---
    k_gemm_wmma<HIDF><<<gemmGrd1, gemmBlk, 0, stream>>>(x, W1, bufA);
    k_init_agg<HIDF><<<(nElemH + T - 1) / T, T, 0, stream>>>(bufA, dinv, b1, bufB, N);
    k_scatter<HIDF><<<scatBlocksH, T, 0, stream>>>(esrc, edst, dinv, bufA, bufB, E);
    k_relu4<<<(nElemH / 4 + T - 1) / T, T, 0, stream>>>(bufB, nElemH / 4);

    // --- layer 2: h2 = relu(agg(h1 @ W2) + b2) ---
    k_gemm_wmma<HIDF><<<gemmGrd1, gemmBlk, 0, stream>>>(bufB, W2, bufA);
    k_init_agg<HIDF><<<(nElemH + T - 1) / T, T, 0, stream>>>(bufA, dinv, b2, bufB, N);
    k_scatter<HIDF><<<scatBlocksH, T, 0, stream>>>(esrc, edst, dinv, bufA, bufB, E);
    k_relu4<<<(nElemH / 4 + T - 1) / T, T, 0, stream>>>(bufB, nElemH / 4);

    // --- layer 3: out = agg(h2 @ W3) + b3 (no relu), directly into d_out ---
    k_gemm_wmma<OUTF><<<gemmGrd3, gemmBlk, 0, stream>>>(bufB, W3, bufA);
    k_init_agg<OUTF><<<(nElemO + T - 1) / T, T, 0, stream>>>(bufA, dinv, b3, out, N);
    k_scatter<OUTF><<<scatBlocksO, T, 0, stream>>>(esrc, edst, dinv, bufA, out, E);
}